// PARE_Net_22832046146326
// MI455X (gfx1250) — compile-verified
//
#include <hip/hip_runtime.h>
#include <hip/hip_bf16.h>

// Segment-mean of two ragged f32 feature sets into [16, 384] (coarse 256 | fine 128).
// Bandwidth-bound streaming; the reduction runs on the matrix pipe via
// V_WMMA_F32_16X16X4_F32 with an all-ones A (full f32 precision; the column sum is
// invariant to the K<->lane permutation of the B layout, only N=lane&15 matters).

typedef __attribute__((ext_vector_type(2))) float v2f;
typedef __attribute__((ext_vector_type(8))) float v8f;

#define NSEG 16
#define OUT_STRIDE 384

__global__ __launch_bounds__(256)
void zero_out_kernel(float* __restrict__ p, int n) {
    int i = blockIdx.x * blockDim.x + threadIdx.x;
    if (i < n) p[i] = 0.0f;
}

// grid.x = NSEG * (D/16) * CHUNKS, block = 256 threads (8 waves).
// Each wave owns a 16-column strip of one segment and walks 4-row tiles at a
// wave-unique stride; WMMA column-sums each 4x16 tile into a v8f accumulator.
template <int D, int CHUNKS>
__global__ __launch_bounds__(256)
void seg_sum_wmma_kernel(const float* __restrict__ feats,
                         const int* __restrict__ lengths,
                         float* __restrict__ acc,   // d_out base (f32 accumulators)
                         int colOffset)
{
    constexpr int CB    = D / 16;          // column blocks (8 or 16, power of two)
    constexpr int TSTEP = CHUNKS * 8;      // waves striping one (seg, colBlock)
    constexpr long long STEP = (long long)TSTEP * 4 * D;  // elements per t-stride

    const int bid      = blockIdx.x;
    const int seg      = bid & (NSEG - 1);
    const int tmp      = bid >> 4;
    const int colBlock = tmp & (CB - 1);
    const int chunk    = tmp / CB;

    // segment bounds: 16-entry scalar prefix (seg is scalar -> s_loads)
    int segStart = 0;
    for (int j = 0; j < seg; ++j) segStart += lengths[j];
    const int segLen = lengths[seg];

    const int wave = __builtin_amdgcn_readfirstlane(threadIdx.x >> 5); // scalar
    const int lane = threadIdx.x & 31;
    const int half = lane >> 4;            // 0: K rows {0,1}, 1: K rows {2,3}
    const int n    = lane & 15;            // N position (layout-stable)
    const int col  = colBlock * 16 + n;

    const int tbeg      = chunk * 8 + wave;    // scalar
    const int fullTiles = segLen >> 2;         // scalar

    const float* base = feats + (long long)segStart * D + col;

    v8f c0 = {};
    v8f c1 = {};
    v2f a;  a.x = 1.0f;  a.y = 1.0f;       // all-ones A => D[m][n] = sum_k B[k][n] + C[m][n]

    int t = tbeg;
    const float* p = base + (long long)(t * 4 + half * 2) * D;

    // main loop: two full tiles per iteration, all offsets are immediates
    for (; t + TSTEP < fullTiles; t += 2 * TSTEP) {
        v2f b0;  b0.x = p[0];        b0.y = p[D];
        v2f b1;  b1.x = p[STEP];     b1.y = p[STEP + D];
        __builtin_prefetch(p + 2 * STEP, 0, 0);   // global_prefetch_b8, next pair
        c0 = __builtin_amdgcn_wmma_f32_16x16x4_f32(false, a, false, b0,
                                                   (short)0, c0, false, false);
        c1 = __builtin_amdgcn_wmma_f32_16x16x4_f32(false, a, false, b1,
                                                   (short)0, c1, false, false);
        p += 2 * STEP;
    }
    // at most one leftover full tile
    for (; t < fullTiles; t += TSTEP) {
        v2f b;  b.x = p[0];  b.y = p[D];
        c0 = __builtin_amdgcn_wmma_f32_16x16x4_f32(false, a, false, b,
                                                   (short)0, c0, false, false);
        p += STEP;
    }
    // partial tail tile: exactly one wave lands with t == fullTiles
    if ((segLen & 3) && (t == fullTiles)) {
        const int r0 = fullTiles * 4 + half * 2;
        v2f b;
        b.x = (r0     < segLen) ? p[0] : 0.0f;
        b.y = (r0 + 1 < segLen) ? p[D] : 0.0f;
        c1 = __builtin_amdgcn_wmma_f32_16x16x4_f32(false, a, false, b,
                                                   (short)0, c1, false, false);
    }

    v8f cs = c0 + c1;
    // C/D layout: VGPR0, lanes 0..15 hold (M=0, N=lane) -> the column sum.
    __shared__ float red[16];
    if (threadIdx.x < 16) red[threadIdx.x] = 0.0f;
    __syncthreads();
    if (half == 0) atomicAdd(&red[n], cs[0]);          // ds_add_f32 (8 waves collapse)
    __syncthreads();
    if (threadIdx.x < 16) {
        atomicAdd(&acc[seg * OUT_STRIDE + colOffset + colBlock * 16 + threadIdx.x],
                  red[threadIdx.x]);                   // global_atomic_add_f32
    }
}

__global__ __launch_bounds__(256)
void finalize_kernel(float* __restrict__ out,
                     const int* __restrict__ lenC,
                     const int* __restrict__ lenF)
{
    int i = blockIdx.x * blockDim.x + threadIdx.x;
    if (i >= NSEG * OUT_STRIDE) return;
    int b = i / OUT_STRIDE;
    int d = i - b * OUT_STRIDE;
    float len = (d < 256) ? (float)lenC[b] : (float)lenF[b];
    out[i] = out[i] / len;
}

extern "C" void kernel_launch(void* const* d_in, const int* in_sizes, int n_in,
                              void* d_out, int out_size, void* d_ws, size_t ws_size,
                              hipStream_t stream) {
    (void)in_sizes; (void)n_in; (void)d_ws; (void)ws_size; (void)out_size;

    const float* feats_f   = (const float*)d_in[0];   // [524288, 128]
    const float* feats_c   = (const float*)d_in[1];   // [32768, 256]
    const int*   lengths_f = (const int*)d_in[2];     // [16]
    const int*   lengths_c = (const int*)d_in[3];     // [16]
    float*       out       = (float*)d_out;           // [16, 384]

    const int outElems = NSEG * OUT_STRIDE;           // 6144

    // 1) zero accumulators (d_out is the accumulator)
    zero_out_kernel<<<(outElems + 255) / 256, 256, 0, stream>>>(out, outElems);

    // 2) coarse: D=256 -> 16 col-blocks, 8 chunks -> 2048 blocks
    seg_sum_wmma_kernel<256, 8>
        <<<NSEG * (256 / 16) * 8, 256, 0, stream>>>(feats_c, lengths_c, out, 0);

    // 3) fine: D=128 -> 8 col-blocks, 32 chunks -> 4096 blocks
    seg_sum_wmma_kernel<128, 32>
        <<<NSEG * (128 / 16) * 32, 256, 0, stream>>>(feats_f, lengths_f, out, 256);

    // 4) divide by segment lengths
    finalize_kernel<<<(outElems + 255) / 256, 256, 0, stream>>>(out, lengths_c, lengths_f);
}